// AdjacencyGenerator_86638080295597
// MI455X (gfx1250) — compile-verified
//
#include <hip/hip_runtime.h>
#include <math.h>

typedef float v2f __attribute__((ext_vector_type(2)));
typedef float v8f __attribute__((ext_vector_type(8)));

#define D       128
#define LN_EPS  1e-5f
#define SM_EPS  1e-16f

#if defined(__gfx1250__) && __has_builtin(__builtin_amdgcn_wmma_f32_16x16x4_f32)
#define USE_WMMA_F32 1
#else
#define USE_WMMA_F32 0
#endif

// ---------------------------------------------------------------------------
// Node-side GEMM: out[r,n] = sum_k in[r,k]*W[n,k] + bias[n]  (optional relu)
// One block = 16 rows x 128 cols. 8 waves, wave w owns cols [16w,16w+16).
// A tile staged in LDS; V_WMMA_F32_16X16X4_F32, K stepped by 4.
// Requires nrows % 16 == 0 (N = 50000 = 3125*16).
// ---------------------------------------------------------------------------
__global__ __launch_bounds__(256)
void node_linear_wmma(const float* __restrict__ in, const float* __restrict__ W,
                      const float* __restrict__ bias, float* __restrict__ out,
                      int do_relu) {
    __shared__ float tile[16 * D];                  // 8 KB
    const int t = threadIdx.x;
    const long base = (long)blockIdx.x * 16 * D;

    // cooperative load of the 16x128 input tile (512 float4 by 256 threads)
    const float4* gin = (const float4*)(in + base);
    float4* stile = (float4*)tile;
    stile[t]       = gin[t];
    stile[t + 256] = gin[t + 256];
    __syncthreads();

    const int lane = t & 31;
    const int wave = t >> 5;
    const int lr = lane & 15;                       // row-within-16 (A: M, B: N)
    const int h  = lane >> 4;                       // lane half selects K pair
    const int n  = wave * 16 + lr;                  // output column for B loads

    const float* arow = tile + lr * D;              // A: input row (LDS)
    const float* wrow = W + (long)n * D;            // B: W row n (W.T column)

    v8f acc = {};
#if USE_WMMA_F32
    #pragma unroll
    for (int k0 = 0; k0 < D; k0 += 4) {
        const int kk = k0 + 2 * h;                  // ISA f32 A/B layout: lanes
        v2f a = *(const v2f*)(arow + kk);           // 0-15 -> K={k0,k0+1},
        v2f b = *(const v2f*)(wrow + kk);           // 16-31 -> K={k0+2,k0+3}
        acc = __builtin_amdgcn_wmma_f32_16x16x4_f32(
            /*neg_a=*/false, a, /*neg_b=*/false, b,
            /*c_mod=*/(short)0, acc, /*reuse_a=*/false, /*reuse_b=*/false);
    }
#else
    // VALU fallback (compile safety only)
    #pragma unroll
    for (int v = 0; v < 8; ++v) {
        const int mm = v + 8 * h;
        float s = 0.f;
        for (int k = 0; k < D; ++k) s += tile[mm * D + k] * wrow[k];
        acc[v] = s;
    }
#endif
    const float bn = bias ? bias[n] : 0.f;
    #pragma unroll
    for (int v = 0; v < 8; ++v) {                   // D layout: VGPR v -> M=v+8h
        const int m = v + 8 * h;
        float r = acc[v] + bn;
        if (do_relu) r = fmaxf(r, 0.f);
        out[base + (long)m * D + n] = r;
    }
}

// ---------------------------------------------------------------------------
// Collapse the activation-free MLP: tmpW = W2@W1, t2 = W2@b1 + b2
// ---------------------------------------------------------------------------
__global__ void combine1(const float* __restrict__ W1, const float* __restrict__ b1,
                         const float* __restrict__ W2, const float* __restrict__ b2,
                         float* __restrict__ tmpW, float* __restrict__ t2) {
    const int tid = blockIdx.x * blockDim.x + threadIdx.x;
    if (tid < 256 * 128) {
        const int i = tid / 128, j = tid % 128;
        float acc = 0.f;
        for (int k = 0; k < 512; ++k) acc += W2[i * 512 + k] * W1[k * 128 + j];
        tmpW[tid] = acc;
    } else if (tid < 256 * 128 + 256) {
        const int i = tid - 256 * 128;
        float acc = b2[i];
        for (int k = 0; k < 512; ++k) acc += W2[i * 512 + k] * b1[k];
        t2[i] = acc;
    }
}

// Wc = W3@tmpW, bc = W3@t2 + b3
__global__ void combine2(const float* __restrict__ W3, const float* __restrict__ b3,
                         const float* __restrict__ tmpW, const float* __restrict__ t2,
                         float* __restrict__ Wc, float* __restrict__ bc) {
    const int tid = blockIdx.x * blockDim.x + threadIdx.x;
    if (tid < 128 * 128) {
        const int i = tid / 128, j = tid % 128;
        float acc = 0.f;
        for (int k = 0; k < 256; ++k) acc += W3[i * 256 + k] * tmpW[k * 128 + j];
        Wc[tid] = acc;
    } else if (tid < 128 * 128 + 128) {
        const int i = tid - 128 * 128;
        float acc = b3[i];
        for (int k = 0; k < 256; ++k) acc += W3[i * 256 + k] * t2[k];
        bc[i] = acc;
    }
}

// ---------------------------------------------------------------------------
// Segment-softmax scaffolding (monotone float->uint key for atomicMax)
// ---------------------------------------------------------------------------
__device__ __forceinline__ unsigned f2key(float f) {
    const unsigned u = __float_as_uint(f);
    return ((int)u >= 0) ? (u | 0x80000000u) : ~u;
}
__device__ __forceinline__ float key2f(unsigned k) {
    const unsigned u = (k & 0x80000000u) ? (k ^ 0x80000000u) : ~k;
    return __uint_as_float(u);
}

__global__ void init_seg(unsigned* __restrict__ m, float* __restrict__ s, int N) {
    const int i = blockIdx.x * blockDim.x + threadIdx.x;
    if (i < N) { m[i] = 0u; s[i] = 0.f; }
}

// alpha[e] = dot(Q[src],K[dst]); segment max into m[dst]. Warp per edge.
__global__ __launch_bounds__(256)
void edge_alpha_max(const int* __restrict__ src, const int* __restrict__ dst,
                    const float* __restrict__ Q, const float* __restrict__ Kb,
                    float* __restrict__ alpha, unsigned* __restrict__ m, int E) {
    const int e = (blockIdx.x * blockDim.x + threadIdx.x) >> 5;
    const int lane = threadIdx.x & 31;
    if (e >= E) return;
    const int si = src[e], di = dst[e];
    __builtin_prefetch(Kb + (long)di * D + lane * 4, 0, 1);
    const float4 q = *(const float4*)(Q  + (long)si * D + lane * 4);
    const float4 k = *(const float4*)(Kb + (long)di * D + lane * 4);
    float p = q.x * k.x + q.y * k.y + q.z * k.z + q.w * k.w;
    #pragma unroll
    for (int off = 16; off; off >>= 1) p += __shfl_xor(p, off);
    if (lane == 0) {
        alpha[e] = p;
        atomicMax(&m[di], f2key(p));
    }
}

// e_exp = exp(alpha - max[dst]); segment sum into s[dst]. Thread per edge.
__global__ void edge_exp_sum(const int* __restrict__ dst, const unsigned* __restrict__ m,
                             float* __restrict__ alpha, float* __restrict__ s, int E) {
    const int e = blockIdx.x * blockDim.x + threadIdx.x;
    if (e >= E) return;
    const int d = dst[e];
    const float ex = expf(alpha[e] - key2f(m[d]));
    alpha[e] = ex;
    atomicAdd(&s[d], ex);
}

// ---------------------------------------------------------------------------
// Fused tail per edge (warp per edge, lane owns 4 channels):
//   a      = e/(s+eps)
//   query  = LN(x[dst] + a*Vr[dst]; ln_g, ln_b)
//   z      = a*U[dst] + bc + query
//   out[e] = dot(LN(z; fn_g, fn_b), Wvec) + bvec
// ---------------------------------------------------------------------------
__device__ __forceinline__ void warp_red2(float& a, float& b) {
    #pragma unroll
    for (int off = 16; off; off >>= 1) { a += __shfl_xor(a, off); b += __shfl_xor(b, off); }
}

__global__ __launch_bounds__(256)
void edge_final(const int* __restrict__ dst, const float* __restrict__ x,
                const float* __restrict__ Vr, const float* __restrict__ U,
                const float* __restrict__ e_exp, const float* __restrict__ ssum,
                const float* __restrict__ bc,
                const float* __restrict__ lng, const float* __restrict__ lnb,
                const float* __restrict__ fng, const float* __restrict__ fnb,
                const float* __restrict__ Wvec, const float* __restrict__ bvec,
                float* __restrict__ out, int E) {
    const int e = (blockIdx.x * blockDim.x + threadIdx.x) >> 5;
    const int lane = threadIdx.x & 31;
    if (e >= E) return;
    const int d = dst[e];
    const long rb = (long)d * D + lane * 4;
    __builtin_prefetch(U + rb, 0, 1);
    __builtin_prefetch(Vr + rb, 0, 1);

    const float a = e_exp[e] / (ssum[d] + SM_EPS);
    const float4 xj = *(const float4*)(x  + rb);
    const float4 vr = *(const float4*)(Vr + rb);
    const float4 uu = *(const float4*)(U  + rb);
    const int c = lane * 4;

    // t = x_j + a*relu(V)  -> LayerNorm 1
    float4 tv;
    tv.x = fmaf(a, vr.x, xj.x); tv.y = fmaf(a, vr.y, xj.y);
    tv.z = fmaf(a, vr.z, xj.z); tv.w = fmaf(a, vr.w, xj.w);
    float sm = tv.x + tv.y + tv.z + tv.w;
    float sq = tv.x * tv.x + tv.y * tv.y + tv.z * tv.z + tv.w * tv.w;
    warp_red2(sm, sq);
    float mu  = sm * (1.f / D);
    float inv = rsqrtf(sq * (1.f / D) - mu * mu + LN_EPS);
    const float4 g1 = *(const float4*)(lng + c);
    const float4 b1 = *(const float4*)(lnb + c);
    float4 qy;
    qy.x = (tv.x - mu) * inv * g1.x + b1.x;
    qy.y = (tv.y - mu) * inv * g1.y + b1.y;
    qy.z = (tv.z - mu) * inv * g1.z + b1.z;
    qy.w = (tv.w - mu) * inv * g1.w + b1.w;

    // z = a*U + bc + query  -> LayerNorm 2
    const float4 bcv = *(const float4*)(bc + c);
    float4 z;
    z.x = fmaf(a, uu.x, bcv.x) + qy.x;
    z.y = fmaf(a, uu.y, bcv.y) + qy.y;
    z.z = fmaf(a, uu.z, bcv.z) + qy.z;
    z.w = fmaf(a, uu.w, bcv.w) + qy.w;
    sm = z.x + z.y + z.z + z.w;
    sq = z.x * z.x + z.y * z.y + z.z * z.z + z.w * z.w;
    warp_red2(sm, sq);
    mu  = sm * (1.f / D);
    inv = rsqrtf(sq * (1.f / D) - mu * mu + LN_EPS);
    const float4 g2 = *(const float4*)(fng + c);
    const float4 b2 = *(const float4*)(fnb + c);
    const float4 wv = *(const float4*)(Wvec + c);

    float part = ((z.x - mu) * inv * g2.x + b2.x) * wv.x
               + ((z.y - mu) * inv * g2.y + b2.y) * wv.y
               + ((z.z - mu) * inv * g2.z + b2.z) * wv.z
               + ((z.w - mu) * inv * g2.w + b2.w) * wv.w;
    #pragma unroll
    for (int off = 16; off; off >>= 1) part += __shfl_xor(part, off);
    if (lane == 0) out[e] = part + bvec[0];
}

// ---------------------------------------------------------------------------
static inline char* carve(char*& p, size_t bytes) {
    char* r = p;
    p += (bytes + 255) & ~(size_t)255;
    return r;
}

extern "C" void kernel_launch(void* const* d_in, const int* in_sizes, int n_in,
                              void* d_out, int out_size, void* d_ws, size_t ws_size,
                              hipStream_t stream) {
    const int* edge = (const int*)d_in[0];
    const int  E = in_sizes[0] / 2;
    const float* x = (const float*)d_in[1];
    const int  N = in_sizes[1] / D;
    const int  L = in_sizes[2] / (D * D);     // only last layer is live
    const int  li = L - 1;

    const float* Wq  = (const float*)d_in[2]  + (size_t)li * D * D;
    const float* bq  = (const float*)d_in[3]  + (size_t)li * D;
    const float* Wk  = (const float*)d_in[4]  + (size_t)li * D * D;
    const float* bk  = (const float*)d_in[5]  + (size_t)li * D;
    const float* Wv  = (const float*)d_in[6]  + (size_t)li * D * D;
    const float* bv  = (const float*)d_in[7]  + (size_t)li * D;
    const float* lng = (const float*)d_in[8]  + (size_t)li * D;
    const float* lnb = (const float*)d_in[9]  + (size_t)li * D;
    const float* W1  = (const float*)d_in[10];
    const float* b1  = (const float*)d_in[11];
    const float* W2  = (const float*)d_in[12];
    const float* b2  = (const float*)d_in[13];
    const float* W3  = (const float*)d_in[14];
    const float* b3  = (const float*)d_in[15];
    const float* Wvec = (const float*)d_in[16];
    const float* bvec = (const float*)d_in[17];
    const float* fng = (const float*)d_in[18];
    const float* fnb = (const float*)d_in[19];

    char* p = (char*)d_ws;
    float*    B0    = (float*)   carve(p, (size_t)N * D * 4);  // Q, later relu(V)
    float*    B1    = (float*)   carve(p, (size_t)N * D * 4);  // K, later U
    float*    alpha = (float*)   carve(p, (size_t)E * 4);
    unsigned* mseg  = (unsigned*)carve(p, (size_t)N * 4);
    float*    sseg  = (float*)   carve(p, (size_t)N * 4);
    float*    tmpW  = (float*)   carve(p, 256 * 128 * 4);
    float*    t2    = (float*)   carve(p, 256 * 4);
    float*    Wc    = (float*)   carve(p, 128 * 128 * 4);
    float*    bc    = (float*)   carve(p, 128 * 4);

    const int tiles = N / 16;                 // N = 50000 = 3125 * 16
    const int ewarp = (E + 7) / 8;            // warp-per-edge blocks (256 thr)

    // Fold MLP into single affine map
    combine1<<<(256 * 128 + 256 + 255) / 256, 256, 0, stream>>>(W1, b1, W2, b2, tmpW, t2);
    combine2<<<(128 * 128 + 128 + 255) / 256, 256, 0, stream>>>(W3, b3, tmpW, t2, Wc, bc);

    // Per-node projections (WMMA): Q, K
    node_linear_wmma<<<tiles, 256, 0, stream>>>(x, Wq, bq, B0, 0);
    node_linear_wmma<<<tiles, 256, 0, stream>>>(x, Wk, bk, B1, 0);
    init_seg<<<(N + 255) / 256, 256, 0, stream>>>(mseg, sseg, N);

    // alpha + segment max
    edge_alpha_max<<<ewarp, 256, 0, stream>>>(edge, edge + E, B0, B1, alpha, mseg, E);

    // relu(V) then U = relu(V) @ Wc.T  (reuse B0/B1, Q/K are dead now)
    node_linear_wmma<<<tiles, 256, 0, stream>>>(x, Wv, bv, B0, 1);
    node_linear_wmma<<<tiles, 256, 0, stream>>>(B0, Wc, nullptr, B1, 0);

    // exp + segment sum, then fused LN/LN/dot tail
    edge_exp_sum<<<(E + 255) / 256, 256, 0, stream>>>(edge + E, mseg, alpha, sseg, E);
    edge_final<<<ewarp, 256, 0, stream>>>(edge + E, x, B0, B1, alpha, sseg, bc,
                                          lng, lnb, fng, fnb, Wvec, bvec,
                                          (float*)d_out, E);
}